// AFM_25348896981773
// MI455X (gfx1250) — compile-verified
//
#include <hip/hip_runtime.h>
#include <hip/hip_bf16.h>

typedef __attribute__((ext_vector_type(2))) float v2f;
typedef __attribute__((ext_vector_type(8))) float v8f;

#define NF     50
#define ND     16
#define NA     32
#define NPAIRS 1225      // 50*49/2
#define NTILES 77        // ceil(1225/16), 77*16 = 1232
#define BLOCK  128       // 4 wave32

// ---------------------------------------------------------------------------
// Kernel 1: fill triu(k=1) pair table, packed as (row | col<<8) u16.
// ---------------------------------------------------------------------------
__global__ void afm_init_pairs(unsigned short* __restrict__ pairs) {
    int p = blockIdx.x * blockDim.x + threadIdx.x;
    if (p >= NPAIRS) return;
    int r = 0, rem = p;
    while (rem >= (NF - 1 - r)) { rem -= (NF - 1 - r); ++r; }
    pairs[p] = (unsigned short)(r | ((r + 1 + rem) << 8));
}

// ---------------------------------------------------------------------------
// Kernel 2: one workgroup per batch row. WMMA f32 16x16x4 for the
// (1225 x 16) x (16 x 32) attention GEMM, then softmax + pooling in LDS.
// ---------------------------------------------------------------------------
__global__ __launch_bounds__(BLOCK) void afm_main(
    const float* __restrict__ feat,   // (B, 50, 16)
    const float* __restrict__ W,      // (16, 32)
    const float* __restrict__ h,      // (32)
    const float* __restrict__ pvec,   // (16)
    const unsigned short* __restrict__ pairs,
    float* __restrict__ out)          // (B, 1)
{
    __shared__ __align__(16) float se[NF * ND];        // staged embeddings, 800 f32
    __shared__ __align__(16) float sc[NTILES * 16];    // scores -> exp(scores), 7 pad slots
    __shared__ float pred[BLOCK];                      // pooled partials [8][16]
    __shared__ float sred[4];                          // per-wave max
    __shared__ float sred2[4];                         // per-wave sum
    __shared__ __align__(4) unsigned short lpair[NTILES * 16];

    const int tid  = threadIdx.x;
    const int lane = tid & 31;
    const int wid  = tid >> 5;
    const int b    = blockIdx.x;

    // ---- stage embeddings (b128) + packed pair table (b32 words) into LDS ----
    {
        const float4* fb4 = (const float4*)(feat + (size_t)b * (NF * ND));
        float4* se4 = (float4*)se;
        for (int i = tid; i < (NF * ND) / 4; i += BLOCK) se4[i] = fb4[i];

        const unsigned int* p32 = (const unsigned int*)pairs;
        unsigned int* lp32 = (unsigned int*)lpair;
        for (int i = tid; i < (NTILES * 16) / 2; i += BLOCK) lp32[i] = p32[i];
    }

    // ---- W in B-operand register layout (K x N = 4 x 16 per WMMA step) ----
    // lanes 0-15: VGPR0=K(4j+0), VGPR1=K(4j+1);  lanes 16-31: K(4j+2), K(4j+3)
    const int nlo  = lane & 15;
    const int krow = (lane < 16) ? 0 : 2;
    v2f bw0[4], bw1[4];
#pragma unroll
    for (int j = 0; j < 4; ++j) {
        const int k0 = 4 * j + krow;
        bw0[j].x = W[(k0    ) * NA + nlo];
        bw0[j].y = W[(k0 + 1) * NA + nlo];
        bw1[j].x = W[(k0    ) * NA + 16 + nlo];
        bw1[j].y = W[(k0 + 1) * NA + 16 + nlo];
    }
    const float h0 = h[nlo];        // for N-half 0 (cols 0..15)
    const float h1 = h[nlo + 16];   // for N-half 1 (cols 16..31)

    __syncthreads();

    // ---- scores: 77 tiles of 16 pairs, K=16 via 4 chained 16x16x4 WMMAs ----
    const int m     = lane & 15;
    const int khalf = krow >> 1;    // 0 (lanes 0-15) or 1 (lanes 16-31)
    for (int t = wid; t < NTILES; t += (BLOCK / 32)) {
        const int p  = t * 16 + m;
        const int pp = p < NPAIRS ? p : NPAIRS - 1;   // clamp: padded rows are
                                                      // computed but never read
        const unsigned int pc = lpair[pp];            // one ds_load_u16
        const float2* er2 = (const float2*)(se + (pc & 0xffu) * ND);
        const float2* ec2 = (const float2*)(se + (pc >> 8)    * ND);

        // branch-free A fragments: lane holds inter[p][4j+krow], [4j+krow+1]
        v2f a[4];
#pragma unroll
        for (int j = 0; j < 4; ++j) {
            const float2 u = er2[2 * j + khalf];      // ds_load_b64 (2addr-paired)
            const float2 v = ec2[2 * j + khalf];
            a[j].x = u.x * v.x;
            a[j].y = u.y * v.y;
        }

        v8f acc0 = {}; v8f acc1 = {};
#pragma unroll
        for (int j = 0; j < 4; ++j) {
            acc0 = __builtin_amdgcn_wmma_f32_16x16x4_f32(false, a[j], false, bw0[j],
                                                         (short)0, acc0, false, false);
            acc1 = __builtin_amdgcn_wmma_f32_16x16x4_f32(false, a[j], false, bw1[j],
                                                         (short)0, acc1, false, false);
        }

        // relu + dot with h; batched 4-stage butterfly over the 16 N-lanes of
        // each half-wave (8 independent bpermutes per stage -> one wait each)
        float s[8];
#pragma unroll
        for (int i = 0; i < 8; ++i)
            s[i] = fmaxf(acc0[i], 0.0f) * h0 + fmaxf(acc1[i], 0.0f) * h1;
#pragma unroll
        for (int msk = 1; msk <= 8; msk <<= 1) {
#pragma unroll
            for (int i = 0; i < 8; ++i) s[i] += __shfl_xor(s[i], msk);
        }
        // lanes 0 / 16 hold rows t*16+0..7 / t*16+8..15: two b128 stores, no guard
        if ((lane & 15) == 0) {
            const int base = t * 16 + ((lane < 16) ? 0 : 8);
            *(float4*)(sc + base)     = float4{s[0], s[1], s[2], s[3]};
            *(float4*)(sc + base + 4) = float4{s[4], s[5], s[6], s[7]};
        }
    }
    __syncthreads();

    // ---- softmax over 1225 scores ----
    float lmax = -3.402823466e38f;
    for (int p = tid; p < NPAIRS; p += BLOCK) lmax = fmaxf(lmax, sc[p]);
#pragma unroll
    for (int msk = 16; msk >= 1; msk >>= 1) lmax = fmaxf(lmax, __shfl_xor(lmax, msk));
    if (lane == 0) sred[wid] = lmax;
    __syncthreads();
    const float gmax = fmaxf(fmaxf(sred[0], sred[1]), fmaxf(sred[2], sred[3]));

    float lsum = 0.0f;
    for (int p = tid; p < NPAIRS; p += BLOCK) {
        const float ev = __expf(sc[p] - gmax);
        sc[p] = ev;                 // overwrite with unnormalized weights
        lsum += ev;
    }
#pragma unroll
    for (int msk = 16; msk >= 1; msk >>= 1) lsum += __shfl_xor(lsum, msk);
    if (lane == 0) sred2[wid] = lsum;
    __syncthreads();
    const float gsum = sred2[0] + sred2[1] + sred2[2] + sred2[3];

    // ---- pooling: pooled[d] = sum_p w[p] * e[r][d] * e[c][d]  (recompute) ----
    const int d = tid & 15;
    const int g = tid >> 4;         // 0..7 pair groups
    float pd = 0.0f;
    for (int p = g; p < NPAIRS; p += 8) {
        const float w = sc[p];
        const unsigned int pc = lpair[p];
        pd += w * se[(pc & 0xffu) * ND + d] * se[(pc >> 8) * ND + d];
    }
    pred[g * 16 + d] = pd;
    __syncthreads();

    if (tid < 16) {
        float tot = 0.0f;
#pragma unroll
        for (int g2 = 0; g2 < 8; ++g2) tot += pred[g2 * 16 + tid];
        float val = tot * pvec[tid];
        val += __shfl_xor(val, 1);
        val += __shfl_xor(val, 2);
        val += __shfl_xor(val, 4);
        val += __shfl_xor(val, 8);
        if (tid == 0) out[b] = val / gsum;   // fold softmax normalization in once
    }
}

// ---------------------------------------------------------------------------
extern "C" void kernel_launch(void* const* d_in, const int* in_sizes, int n_in,
                              void* d_out, int out_size, void* d_ws, size_t ws_size,
                              hipStream_t stream) {
    const float* feat = (const float*)d_in[0];   // (B, 50, 16) f32
    const float* W    = (const float*)d_in[1];   // (16, 32)    f32
    const float* h    = (const float*)d_in[2];   // (32)        f32
    const float* pvec = (const float*)d_in[3];   // (16)        f32
    float*       out  = (float*)d_out;           // (B, 1)      f32

    unsigned short* pairs = (unsigned short*)d_ws;   // 1232 u16 = 2464 B of ws

    const int B = in_sizes[0] / (NF * ND);           // 4096

    afm_init_pairs<<<(NPAIRS + 255) / 256, 256, 0, stream>>>(pairs);
    afm_main<<<B, BLOCK, 0, stream>>>(feat, W, h, pvec, pairs, out);
}